// TabR_52501680226764
// MI455X (gfx1250) — compile-verified
//
#include <hip/hip_runtime.h>
#include <hip/hip_bf16.h>
#include <math.h>

#define DEV __device__ __forceinline__

typedef __attribute__((ext_vector_type(2))) float v2f;
typedef __attribute__((ext_vector_type(8))) float v8f;

enum { B_ = 1024, D_ = 256, H_ = 512, N_ = 100000, K_ = 96, C_ = 10 };
enum { TPB = 256, BSTR = 520 /* padded LDS stride (floats) to dodge bank conflicts */,
       CHUNK = 64 /* query rows per sim/topk chunk -> 25.6MB sim buffer */ };

DEV v8f zero8() { v8f z = {0.f,0.f,0.f,0.f,0.f,0.f,0.f,0.f}; return z; }

// fp32 matrix core: D = A(16x4) * B(4x16) + C(16x16)
DEV v8f wmma4(v2f a, v2f b, v8f c) {
  return __builtin_amdgcn_wmma_f32_16x16x4_f32(false, a, false, b, (short)0, c,
                                               false, false);
}

// A fragment: rows striped lane%16, K pair selected by lane half (ISA 7.12.2)
DEV v2f load_a(const float* t, int ld, int k, int lane) {
  int row = lane & 15;
  int kk  = k + ((lane >> 4) << 1);
  v2f a; a.x = t[row * ld + kk]; a.y = t[row * ld + kk + 1]; return a;
}

// B fragment from a row-major matrix: B[k][n] = W[k*ldw + n]
DEV v2f load_b_rm(const float* __restrict__ W, int ldw, int k, int colbase, int lane) {
  int col = colbase + (lane & 15);
  int kk  = k + ((lane >> 4) << 1);
  v2f b; b.x = W[(size_t)kk * ldw + col]; b.y = W[(size_t)(kk + 1) * ldw + col];
  return b;
}

// ---------------------------------------------------------------------------
// Fused encoder: keys = (X@W1 + b1)@W2 + b2 per 16-row tile, optional enc out.
// ---------------------------------------------------------------------------
__global__ __launch_bounds__(TPB)
void encode_kernel(const float* __restrict__ X,
                   const float* __restrict__ W1, const float* __restrict__ b1,
                   const float* __restrict__ W2, const float* __restrict__ b2,
                   float* __restrict__ enc_out, float* __restrict__ keys_out)
{
  __shared__ float buf[16 * BSTR];           // stage0: X tile, stage1: enc tile
  const int tid = threadIdx.x, wave = tid >> 5, lane = tid & 31;
  const int rowbase = blockIdx.x * 16;

  for (int i = tid; i < 16 * D_; i += TPB) {
    int r = i >> 8, c = i & (D_ - 1);
    buf[r * BSTR + c] = X[(size_t)(rowbase + r) * D_ + c];
  }
  __syncthreads();

  v8f acc[4]; for (int t = 0; t < 4; ++t) acc[t] = zero8();
  for (int k = 0; k < D_; k += 4) {
    v2f a = load_a(buf, BSTR, k, lane);
    #pragma unroll
    for (int t = 0; t < 4; ++t)
      acc[t] = wmma4(a, load_b_rm(W1, H_, k, wave * 64 + t * 16, lane), acc[t]);
  }
  __syncthreads();                           // all waves done reading X tile
  #pragma unroll
  for (int t = 0; t < 4; ++t) {
    int col = wave * 64 + t * 16 + (lane & 15);
    #pragma unroll
    for (int v = 0; v < 8; ++v) {
      int m = v + ((lane >> 4) << 3);
      float e = acc[t][v] + b1[col];
      buf[m * BSTR + col] = e;
      if (enc_out) enc_out[(size_t)(rowbase + m) * H_ + col] = e;
    }
  }
  __syncthreads();

  v8f acc2[4]; for (int t = 0; t < 4; ++t) acc2[t] = zero8();
  for (int k = 0; k < H_; k += 4) {
    v2f a = load_a(buf, BSTR, k, lane);
    #pragma unroll
    for (int t = 0; t < 4; ++t)
      acc2[t] = wmma4(a, load_b_rm(W2, H_, k, wave * 64 + t * 16, lane), acc2[t]);
  }
  #pragma unroll
  for (int t = 0; t < 4; ++t) {
    int col = wave * 64 + t * 16 + (lane & 15);
    #pragma unroll
    for (int v = 0; v < 8; ++v) {
      int m = v + ((lane >> 4) << 3);
      keys_out[(size_t)(rowbase + m) * H_ + col] = acc2[t][v] + b2[col];
    }
  }
}

// ---------------------------------------------------------------------------
// Row sum-of-squares (for the cdist GEMM expansion).
// ---------------------------------------------------------------------------
__global__ __launch_bounds__(128)
void sumsq_kernel(const float* __restrict__ keys, float* __restrict__ out)
{
  __shared__ float red[128];
  const int r = blockIdx.x, tid = threadIdx.x;
  float s = 0.f;
  for (int c = tid; c < H_; c += 128) { float v = keys[(size_t)r * H_ + c]; s += v * v; }
  red[tid] = s; __syncthreads();
  for (int off = 64; off > 0; off >>= 1) { if (tid < off) red[tid] += red[tid + off]; __syncthreads(); }
  if (tid == 0) out[r] = red[0];
}

// ---------------------------------------------------------------------------
// sim chunk: 16 query rows x 512 candidate cols per block (8 waves x 4 tiles).
// sim = -sqrt(max(q2 + c2 - 2*dot, 0))
// ---------------------------------------------------------------------------
__global__ __launch_bounds__(TPB)
void sim_kernel(const float* __restrict__ x_key, const float* __restrict__ q2,
                const float* __restrict__ ck, const float* __restrict__ c2,
                int row0, float* __restrict__ simc)
{
  __shared__ float xkt[16 * BSTR];
  __shared__ float q2s[16];
  const int tid = threadIdx.x, wave = tid >> 5, lane = tid & 31;
  const int mrow = blockIdx.y * 16;                 // row tile within chunk
  const int colbase = blockIdx.x * 512 + wave * 64;

  for (int i = tid; i < 16 * H_; i += TPB) {
    int r = i >> 9, c = i & (H_ - 1);
    xkt[r * BSTR + c] = x_key[(size_t)(row0 + mrow + r) * H_ + c];
  }
  if (tid < 16) q2s[tid] = q2[row0 + mrow + tid];
  __syncthreads();

  v8f acc[4]; for (int t = 0; t < 4; ++t) acc[t] = zero8();
  const int kk_half = (lane >> 4) << 1;
  for (int k = 0; k < H_; k += 4) {
    v2f a = load_a(xkt, BSTR, k, lane);
    const int kk = k + kk_half;
    #pragma unroll
    for (int t = 0; t < 4; ++t) {
      int n = colbase + t * 16 + (lane & 15);
      int ncl = n < N_ ? n : (N_ - 1);              // clamp: keep EXEC uniform for WMMA
      const float2 p = *reinterpret_cast<const float2*>(ck + (size_t)ncl * H_ + kk);
      v2f b; b.x = p.x; b.y = p.y;                  // B[k][n] = cand_keys[n][k]
      acc[t] = wmma4(a, b, acc[t]);
    }
  }
  #pragma unroll
  for (int t = 0; t < 4; ++t) {
    int n = colbase + t * 16 + (lane & 15);
    if (n < N_) {
      float c2n = c2[n];
      #pragma unroll
      for (int v = 0; v < 8; ++v) {
        int m = v + ((lane >> 4) << 3);
        float d2 = q2s[m] + c2n - 2.f * acc[t][v];
        simc[(size_t)(mrow + m) * N_ + n] = -sqrtf(fmaxf(d2, 0.f));
      }
    }
  }
}

// ---------------------------------------------------------------------------
// Deterministic top-K (K=96) per row via 4-pass radix select + scan compaction.
// ---------------------------------------------------------------------------
DEV unsigned f2u(float f) {
  unsigned b = __float_as_uint(f);
  return (b & 0x80000000u) ? ~b : (b | 0x80000000u);   // order-preserving map
}

__global__ __launch_bounds__(TPB)
void topk_kernel(const float* __restrict__ simc, int row0,
                 int* __restrict__ tidx, float* __restrict__ tsim)
{
  __shared__ unsigned bins[256];
  __shared__ unsigned sh_prefix; __shared__ int sh_want;
  __shared__ int sgt[TPB], seq[TPB];
  const int tid = threadIdx.x;
  const float* row = simc + (size_t)blockIdx.x * N_;
  const size_t obase = (size_t)(row0 + blockIdx.x) * K_;

  unsigned prefix = 0; int want = K_;
  for (int pass = 0; pass < 4; ++pass) {
    const int shift = 24 - 8 * pass;
    const unsigned hmask = (pass == 0) ? 0u : (0xFFFFFFFFu << (shift + 8));
    for (int i = tid; i < 256; i += TPB) bins[i] = 0u;
    __syncthreads();
    for (int n = tid; n < N_; n += TPB) {
      unsigned u = f2u(row[n]);
      if ((u & hmask) == prefix) atomicAdd(&bins[(u >> shift) & 0xFFu], 1u);
    }
    __syncthreads();
    if (tid == 0) {
      int acc = 0, b = 255;
      for (; b >= 0; --b) {
        int c = (int)bins[b];
        if (acc + c >= want || b == 0) break;
        acc += c;
      }
      sh_prefix = prefix | ((unsigned)b << shift);
      sh_want = want - acc;
    }
    __syncthreads();
    prefix = sh_prefix; want = sh_want;
    __syncthreads();
  }
  const unsigned thresh = prefix;     // exact 32-bit K-th largest key
  const int need_eq = want;
  const int n_gt = K_ - need_eq;

  int cgt = 0, ceq = 0;
  for (int n = tid; n < N_; n += TPB) {
    unsigned u = f2u(row[n]);
    cgt += (u > thresh); ceq += (u == thresh);
  }
  sgt[tid] = cgt; seq[tid] = ceq; __syncthreads();
  for (int off = 1; off < TPB; off <<= 1) {       // Hillis-Steele inclusive scan
    int va = (tid >= off) ? sgt[tid - off] : 0;
    int vb = (tid >= off) ? seq[tid - off] : 0;
    __syncthreads();
    sgt[tid] += va; seq[tid] += vb;
    __syncthreads();
  }
  int pg = sgt[tid] - cgt, pe = seq[tid] - ceq;   // exclusive offsets
  for (int n = tid; n < N_; n += TPB) {
    float f = row[n]; unsigned u = f2u(f);
    if (u > thresh) {
      tidx[obase + pg] = n; tsim[obase + pg] = f; ++pg;
    } else if (u == thresh) {
      if (pe < need_eq) { tidx[obase + n_gt + pe] = n; tsim[obase + n_gt + pe] = f; }
      ++pe;
    }
  }
}

// ---------------------------------------------------------------------------
// Context: one block per query row; softmax over top-96, correction MLP via
// WMMA, attn-weighted sum (fixed lane ownership -> deterministic), + x_enc.
// ---------------------------------------------------------------------------
__global__ __launch_bounds__(TPB)
void context_kernel(const float* __restrict__ x_enc, const float* __restrict__ x_key,
                    const float* __restrict__ ck, const int* __restrict__ labels,
                    const float* __restrict__ E_label,
                    const float* __restrict__ W_t1, const float* __restrict__ b_t1,
                    const float* __restrict__ W_t2,
                    const int* __restrict__ tidx, const float* __restrict__ tsim,
                    float* __restrict__ out)
{
  __shared__ float buf[16 * BSTR];     // diff tile, then relu-hidden tile
  __shared__ float part[2 * H_];       // [col][lane-half] partial sums
  __shared__ float labsum[H_];
  __shared__ float xk[H_];
  __shared__ float attn[K_];
  __shared__ int   idx[K_];

  const int tid = threadIdx.x, wave = tid >> 5, lane = tid & 31;
  const int b = blockIdx.x;

  if (tid < K_) {
    attn[tid] = tsim[(size_t)b * K_ + tid];
    idx[tid]  = tidx[(size_t)b * K_ + tid];
  }
  xk[tid]       = x_key[(size_t)b * H_ + tid];
  xk[tid + TPB] = x_key[(size_t)b * H_ + tid + TPB];
  for (int i = tid; i < 2 * H_; i += TPB) part[i] = 0.f;
  __syncthreads();

  if (tid == 0) {                                  // tiny serial softmax (96 elems)
    float m = attn[0];
    for (int k = 1; k < K_; ++k) m = fmaxf(m, attn[k]);
    float s = 0.f;
    for (int k = 0; k < K_; ++k) { float e = expf(attn[k] - m); attn[k] = e; s += e; }
    float inv = 1.f / s;
    for (int k = 0; k < K_; ++k) attn[k] *= inv;
  }
  __syncthreads();

  for (int col = tid; col < H_; col += TPB) {      // attn-weighted label embeds
    float s = 0.f;
    for (int k = 0; k < K_; ++k)
      s += attn[k] * E_label[(size_t)labels[idx[k]] * H_ + col];
    labsum[col] = s;
  }
  __syncthreads();

  for (int mt = 0; mt < K_ / 16; ++mt) {
    for (int i = tid; i < 16 * H_; i += TPB) {     // diff tile into LDS
      int r = i >> 9, c = i & (H_ - 1);
      buf[r * BSTR + c] = xk[c] - ck[(size_t)idx[mt * 16 + r] * H_ + c];
    }
    __syncthreads();

    v8f acc[4]; for (int t = 0; t < 4; ++t) acc[t] = zero8();
    for (int k = 0; k < H_; k += 4) {
      v2f a = load_a(buf, BSTR, k, lane);
      #pragma unroll
      for (int t = 0; t < 4; ++t)
        acc[t] = wmma4(a, load_b_rm(W_t1, H_, k, wave * 64 + t * 16, lane), acc[t]);
    }
    __syncthreads();                               // done reading diff tile
    #pragma unroll
    for (int t = 0; t < 4; ++t) {
      int col = wave * 64 + t * 16 + (lane & 15);
      #pragma unroll
      for (int v = 0; v < 8; ++v) {
        int m = v + ((lane >> 4) << 3);
        float h = acc[t][v] + b_t1[col];
        buf[m * BSTR + col] = h > 0.f ? h : 0.f;   // relu, reuse buffer
      }
    }
    __syncthreads();

    v8f acc2[4]; for (int t = 0; t < 4; ++t) acc2[t] = zero8();
    for (int k = 0; k < H_; k += 4) {
      v2f a = load_a(buf, BSTR, k, lane);
      #pragma unroll
      for (int t = 0; t < 4; ++t)
        acc2[t] = wmma4(a, load_b_rm(W_t2, H_, k, wave * 64 + t * 16, lane), acc2[t]);
    }
    const int half = lane >> 4;
    #pragma unroll
    for (int t = 0; t < 4; ++t) {                  // attn-weighted sum over 8 rows
      int col = wave * 64 + t * 16 + (lane & 15);
      float s = 0.f;
      #pragma unroll
      for (int v = 0; v < 8; ++v) {
        int m = v + (half << 3);
        s += attn[mt * 16 + m] * acc2[t][v];
      }
      part[col * 2 + half] += s;                   // unique (col,half) per lane
    }
    __syncthreads();
  }

  for (int col = tid; col < H_; col += TPB)
    out[(size_t)b * H_ + col] =
        x_enc[(size_t)b * H_ + col] + labsum[col] + part[col * 2] + part[col * 2 + 1];
}

// ---------------------------------------------------------------------------
extern "C" void kernel_launch(void* const* d_in, const int* in_sizes, int n_in,
                              void* d_out, int out_size, void* d_ws, size_t ws_size,
                              hipStream_t stream)
{
  (void)in_sizes; (void)n_in; (void)out_size; (void)ws_size;
  const float* x       = (const float*)d_in[0];
  const float* candx   = (const float*)d_in[1];
  const int*   labels  = (const int*)d_in[2];
  const float* W_enc   = (const float*)d_in[3];
  const float* b_enc   = (const float*)d_in[4];
  const float* W_key   = (const float*)d_in[5];
  const float* b_key   = (const float*)d_in[6];
  // d_in[7]=W_val, d_in[8]=b_val: dead in the reference (gather result unused)
  const float* E_label = (const float*)d_in[9];
  const float* W_t1    = (const float*)d_in[10];
  const float* b_t1    = (const float*)d_in[11];
  const float* W_t2    = (const float*)d_in[12];
  float* out = (float*)d_out;

  // workspace layout (~235 MB): cand_keys | c2 | x_key | x_enc | q2 | tsim | tidx | sim chunk
  float* ck   = (float*)d_ws;
  float* c2   = ck   + (size_t)N_ * H_;
  float* xkey = c2   + N_;
  float* xenc = xkey + (size_t)B_ * H_;
  float* q2   = xenc + (size_t)B_ * H_;
  float* tsim = q2   + B_;
  int*   tidx = (int*)(tsim + (size_t)B_ * K_);
  float* simc = (float*)(tidx + (size_t)B_ * K_);

  encode_kernel<<<N_ / 16, TPB, 0, stream>>>(candx, W_enc, b_enc, W_key, b_key,
                                             nullptr, ck);
  encode_kernel<<<B_ / 16, TPB, 0, stream>>>(x, W_enc, b_enc, W_key, b_key,
                                             xenc, xkey);
  sumsq_kernel<<<N_, 128, 0, stream>>>(ck, c2);
  sumsq_kernel<<<B_, 128, 0, stream>>>(xkey, q2);

  const int colBlocks = (N_ + 511) / 512;
  for (int chunk = 0; chunk < B_ / CHUNK; ++chunk) {
    const int row0 = chunk * CHUNK;
    sim_kernel<<<dim3(colBlocks, CHUNK / 16), TPB, 0, stream>>>(xkey, q2, ck, c2,
                                                                row0, simc);
    topk_kernel<<<CHUNK, TPB, 0, stream>>>(simc, row0, tidx, tsim);
  }
  context_kernel<<<B_, TPB, 0, stream>>>(xenc, xkey, ck, labels, E_label,
                                         W_t1, b_t1, W_t2, tidx, tsim, out);
}